// GRU_RGCN_75144747811351
// MI455X (gfx1250) — compile-verified
//
#include <hip/hip_runtime.h>
#include <hip/hip_bf16.h>
#include <math.h>

#define T_STEPS 512
#define N_NODES 32
#define D_DIM   300
#define D_PAD   304
#define R_REL   3
#define E_EDGE  64
#define VG_SZ   40000

typedef float v2f __attribute__((ext_vector_type(2)));
typedef float v8f __attribute__((ext_vector_type(8)));

// ---- workspace layout (floats) ----
#define MEM_OFF   0                        // 32*300 = 9600
#define XG_OFF    9600                     // 32*304 = 9728 (gathered x, padded)
#define H_OFF     (9600 + 9728)            // 4 * 32 * 304 = 38912 (3 rel GEMMs + xW0)
#define RELU_OFF  (H_OFF + 38912)          // 304
#define LSE_OFF   (RELU_OFF + 304)         // 16 (padded)
#define WP_OFF    (LSE_OFF + 16)           // 4 * 300 * 304 = 364800 (padded weights)
#define WP_MAT    (D_DIM * D_PAD)          // 91200 floats per matrix

// ---------------------------------------------------------------------------
__global__ void k_zero(float* __restrict__ p, int n) {
    int i = blockIdx.x * blockDim.x + threadIdx.x;
    if (i < n) p[i] = 0.0f;
}

// pack conv_W[0..2] and W0 into [4][300][304] with zeroed pad columns (once per call)
__global__ void k_packW(const float* __restrict__ convW, const float* __restrict__ W0,
                        float* __restrict__ Wp) {
    int i = blockIdx.x * blockDim.x + threadIdx.x;     // 4*300*304 threads
    if (i >= 4 * WP_MAT) return;
    int mat = i / WP_MAT;
    int rem = i - mat * WP_MAT;
    int k = rem / D_PAD, n = rem % D_PAD;
    const float* src = (mat < 3) ? (convW + (size_t)mat * D_DIM * D_DIM) : W0;
    Wp[i] = (n < D_DIM) ? src[(size_t)k * D_DIM + n] : 0.0f;
}

// gather x = X[idx] into padded [32][304] buffer (pad cols zeroed)
__global__ void k_gather(const float* __restrict__ X, const int* __restrict__ idx,
                         float* __restrict__ xg) {
    int i = blockIdx.x * blockDim.x + threadIdx.x;     // 32*304 = 9728 threads
    if (i >= N_NODES * D_PAD) return;
    int n = i / D_PAD, d = i % D_PAD;
    xg[i] = (d < D_DIM) ? X[(size_t)idx[n] * D_DIM + d] : 0.0f;
}

// ---------------------------------------------------------------------------
// h[mat] = x @ W_mat   using V_WMMA_F32_16X16X4_F32.
// grid: (19 N-tiles, 4 matrices), block: 64 (2 waves; wave = M-tile).
// All operand loads are unconditional (both x and W are zero-padded to 304 cols),
// so the 75-iteration K-loop is branch-free: load_b64 / load pair / v_wmma.
__global__ void k_gemm(const float* __restrict__ xg,
                       const float* __restrict__ Wp,
                       float* __restrict__ h) {
    int lane  = threadIdx.x & 31;
    int wave  = threadIdx.x >> 5;          // M-tile 0..1
    int ntile = blockIdx.x;                // 0..18
    int mat   = blockIdx.y;                // 0..3

    int halfsel = (lane >> 4) & 1;         // lanes 16-31 hold K+2,K+3
    int l15     = lane & 15;
    int mr      = wave * 16 + l15;         // A row (0..31)
    int nc      = ntile * 16 + l15;        // B/D column (0..303)

    const float* arow = xg + mr * D_PAD + 2 * halfsel;         // zeros beyond col 299
    const float* bcol = Wp + (size_t)mat * WP_MAT + (size_t)(2 * halfsel) * D_PAD + nc;

    v8f acc = {0.f, 0.f, 0.f, 0.f, 0.f, 0.f, 0.f, 0.f};

    for (int k0 = 0; k0 < D_DIM; k0 += 4) {   // 75 iterations, branch-free
        v2f a, b;
        a.x = arow[k0];
        a.y = arow[k0 + 1];
        b.x = bcol[(size_t)k0 * D_PAD];
        b.y = bcol[(size_t)(k0 + 1) * D_PAD];
        acc = __builtin_amdgcn_wmma_f32_16x16x4_f32(
                  false, a, false, b, (short)0, acc, false, false);
    }

    // D layout: VGPR i -> rows {i, i+8}; lanes 0-15 row i, lanes 16-31 row i+8
    float* hdst = h + (size_t)mat * N_NODES * D_PAD;
    #pragma unroll
    for (int i = 0; i < 8; ++i) {
        int mrow = wave * 16 + i + 8 * halfsel;
        hdst[mrow * D_PAD + nc] = acc[i];
    }
}

// ---------------------------------------------------------------------------
// Single-block per-step update: degree norms, deterministic edge aggregation,
// GRU gate, memory update, relu(mem[0]) extraction.
__global__ void k_update(const float* __restrict__ xg,
                         const int* __restrict__ src,
                         const int* __restrict__ dst,
                         const float* __restrict__ h,
                         const float* __restrict__ Wu,
                         const float* __restrict__ Uu,
                         float* __restrict__ mem,
                         float* __restrict__ relu_row) {
    __shared__ int   deg[R_REL][N_NODES];
    __shared__ float dinv[R_REL][N_NODES];
    __shared__ float gate[D_DIM];
    __shared__ int   ssrc[R_REL * E_EDGE];
    __shared__ int   sdst[R_REL * E_EDGE];

    int tid = threadIdx.x;

    if (tid < R_REL * N_NODES) deg[tid / N_NODES][tid % N_NODES] = 1;  // self loop
    for (int i = tid; i < R_REL * E_EDGE; i += blockDim.x) {
        ssrc[i] = src[i];
        sdst[i] = dst[i];
    }
    __syncthreads();
    for (int i = tid; i < R_REL * E_EDGE; i += blockDim.x)
        atomicAdd(&deg[i / E_EDGE][sdst[i]], 1);                       // int: deterministic
    __syncthreads();
    if (tid < R_REL * N_NODES) {
        int r = tid / N_NODES, n = tid % N_NODES;
        dinv[r][n] = rsqrtf((float)deg[r][n]);                         // deg >= 1 always
    }
    __syncthreads();

    // gate[d] = sigmoid( x_flat . Wu[:,d] + mem_row0 . Uu[:,d] )   (reads OLD mem)
    if (tid < D_DIM) {
        float acc = 0.0f;
        for (int n = 0; n < N_NODES; ++n) {
            const float* xr   = xg + n * D_PAD;
            const float* wcol = Wu + (size_t)(n * D_DIM) * D_DIM + tid;
            for (int dd = 0; dd < D_DIM; ++dd)
                acc = fmaf(xr[dd], wcol[(size_t)dd * D_DIM], acc);
        }
        for (int k = 0; k < D_DIM; ++k)
            acc = fmaf(mem[k], Uu[(size_t)k * D_DIM + tid], acc);
        gate[tid] = 1.0f / (1.0f + __expf(-acc));
    }
    __syncthreads();   // gate done & old mem row0 fully read before overwrite

    // conv + proposed + GRU memory update (deterministic gather over edges)
    for (int i = tid; i < N_NODES * D_DIM; i += blockDim.x) {
        int n = i / D_DIM, d = i % D_DIM;
        float conv = 0.0f;
        #pragma unroll
        for (int r = 0; r < R_REL; ++r) {
            const float* hr = h + (size_t)r * N_NODES * D_PAD;
            float dn = dinv[r][n];
            float c  = hr[n * D_PAD + d] * dn * dn;                    // self loop
            const int* sr = ssrc + r * E_EDGE;
            const int* dr = sdst + r * E_EDGE;
            for (int e = 0; e < E_EDGE; ++e) {
                if (dr[e] == n) {
                    int s = sr[e];
                    c = fmaf(hr[s * D_PAD + d], dinv[r][s] * dn, c);
                }
            }
            conv += c;
        }
        float proposed = conv + h[3 * N_NODES * D_PAD + n * D_PAD + d];  // + x@W0
        float g  = gate[d];
        float mn = g * proposed + (1.0f - g) * mem[i];
        mem[i] = mn;
        if (n == 0) relu_row[d] = fmaxf(mn, 0.0f);
    }
}

// ---------------------------------------------------------------------------
// logits: wave-per-row mat-vec, Wg read exactly once per step (coalesced).
__global__ void k_logits(const float* __restrict__ relu_row,
                         const float* __restrict__ Wg,
                         const float* __restrict__ bg,
                         float* __restrict__ out_row) {
    __shared__ float rv[D_DIM];
    int tid = threadIdx.x;
    for (int i = tid; i < D_DIM; i += blockDim.x) rv[i] = relu_row[i];
    __syncthreads();

    int wave = tid >> 5, lane = tid & 31;
    int v = blockIdx.x * 8 + wave;                 // 5000 blocks * 8 waves = 40000
    const float* wr = Wg + (size_t)v * D_DIM;
    __builtin_prefetch(wr + lane * 8, 0, 0);       // global_prefetch_b8

    float s = 0.0f;
    for (int k = lane; k < D_DIM; k += 32) s = fmaf(rv[k], wr[k], s);
    #pragma unroll
    for (int off = 16; off; off >>= 1) s += __shfl_xor(s, off, 32);
    if (lane == 0) out_row[v] = s + bg[v];
}

// single-block two-pass logsumexp over 40000 logits
__global__ void k_lse(const float* __restrict__ out_row, float* __restrict__ lse) {
    __shared__ float red[1024];
    int tid = threadIdx.x;
    float m = -INFINITY;
    for (int i = tid; i < VG_SZ; i += 1024) m = fmaxf(m, out_row[i]);
    red[tid] = m; __syncthreads();
    for (int s = 512; s; s >>= 1) {
        if (tid < s) red[tid] = fmaxf(red[tid], red[tid + s]);
        __syncthreads();
    }
    float mx = red[0];
    __syncthreads();
    float sum = 0.0f;
    for (int i = tid; i < VG_SZ; i += 1024) sum += __expf(out_row[i] - mx);
    red[tid] = sum; __syncthreads();
    for (int s = 512; s; s >>= 1) {
        if (tid < s) red[tid] += red[tid + s];
        __syncthreads();
    }
    if (tid == 0) *lse = mx + __logf(red[0]);
}

__global__ void k_norm(const float* __restrict__ lse, float* __restrict__ out_row) {
    int i = blockIdx.x * blockDim.x + threadIdx.x;
    if (i < VG_SZ) out_row[i] -= *lse;
}

// second tuple output: zeros([512], int32) == 0x00000000 bits == 0.0f bits
__global__ void k_tail(float* __restrict__ p) {
    p[threadIdx.x] = 0.0f;
}

// ---------------------------------------------------------------------------
extern "C" void kernel_launch(void* const* d_in, const int* in_sizes, int n_in,
                              void* d_out, int out_size, void* d_ws, size_t ws_size,
                              hipStream_t stream) {
    const int*   x_indices = (const int*)d_in[0];   // [512,32]
    const int*   edge_src  = (const int*)d_in[1];   // [512,3,64]
    const int*   edge_dst  = (const int*)d_in[2];   // [512,3,64]
    const float* X         = (const float*)d_in[3]; // [70000,300]
    const float* convW     = (const float*)d_in[4]; // [3,300,300]
    const float* W0        = (const float*)d_in[5]; // [300,300]
    const float* Wu        = (const float*)d_in[6]; // [9600,300]
    const float* Uu        = (const float*)d_in[7]; // [300,300]
    const float* Wg        = (const float*)d_in[8]; // [40000,300]
    const float* bg        = (const float*)d_in[9]; // [40000]

    float* out = (float*)d_out;
    float* ws  = (float*)d_ws;

    float* mem      = ws + MEM_OFF;
    float* xg       = ws + XG_OFF;
    float* h        = ws + H_OFF;
    float* relu_row = ws + RELU_OFF;
    float* lse      = ws + LSE_OFF;
    float* Wp       = ws + WP_OFF;

    // one-time per call: deterministic start state + padded weight pack
    k_zero<<<(N_NODES * D_DIM + 255) / 256, 256, 0, stream>>>(mem, N_NODES * D_DIM);
    k_packW<<<(4 * WP_MAT + 255) / 256, 256, 0, stream>>>(convW, W0, Wp);

    for (int t = 0; t < T_STEPS; ++t) {
        const int* idx_t = x_indices + t * N_NODES;
        const int* src_t = edge_src + t * R_REL * E_EDGE;
        const int* dst_t = edge_dst + t * R_REL * E_EDGE;
        float* out_row   = out + (size_t)t * VG_SZ;

        k_gather<<<(N_NODES * D_PAD) / 256, 256, 0, stream>>>(X, idx_t, xg);
        k_gemm  <<<dim3(19, 4), 64, 0, stream>>>(xg, Wp, h);
        k_update<<<1, 512, 0, stream>>>(xg, src_t, dst_t, h, Wu, Uu, mem, relu_row);
        k_logits<<<VG_SZ / 8, 256, 0, stream>>>(relu_row, Wg, bg, out_row);
        k_lse   <<<1, 1024, 0, stream>>>(out_row, lse);
        k_norm  <<<(VG_SZ + 255) / 256, 256, 0, stream>>>(lse, out_row);
    }

    k_tail<<<1, T_STEPS, 0, stream>>>(out + (size_t)T_STEPS * VG_SZ);
}